// GraphAttentionNetwork_23227183136792
// MI455X (gfx1250) — compile-verified
//
#include <hip/hip_runtime.h>
#include <hip/hip_bf16.h>

#define BATCH 16
#define WN    100     // window length (K of first GEMM, N of last GEMM)
#define FEAT  128     // nodes / feature count (M of both GEMMs)
#define DEMB  200     // embedding dim
#define D2    400     // left+right concatenated
#define ALPHA 0.2f

typedef __attribute__((ext_vector_type(2))) float v2f;
typedef __attribute__((ext_vector_type(8))) float v8f;

// ---------------------------------------------------------------------------
// Kernel 1: LR[b, i, 0:400] = x_b(128x100) @ [W1 | W2](100x400), bias folded
// into the left half.  One wave per 16x16 tile, V_WMMA_F32_16X16X4_F32.
// x_b[i, w] = input[b, w, i].
// ---------------------------------------------------------------------------
__global__ __launch_bounds__(32) void gat_k1_linear(
    const float* __restrict__ inp,     // [B, WN, FEAT]
    const float* __restrict__ lin_w,   // [2*WN, DEMB]
    const float* __restrict__ lin_b,   // [DEMB]
    float* __restrict__ LR)            // [B, FEAT, D2]
{
    const int lane = threadIdx.x;
    const int half = lane >> 4;        // 0: K pair {0,1}, 1: K pair {2,3}
    const int l15  = lane & 15;
    const int n0   = blockIdx.x * 16;  // d-tile in [0, 400)
    const int i0   = blockIdx.y * 16;  // node tile in [0, 128)
    const int b    = blockIdx.z;

    const float* inB = inp + (size_t)b * WN * FEAT;

    // B-matrix column for this lane (constant across K loop)
    const int d       = n0 + l15;
    const int row_off = (d < DEMB) ? 0 : WN;       // W2 = rows 100..199
    const int dc      = (d < DEMB) ? d : d - DEMB;

    v8f c = {};
    for (int k0 = 0; k0 < WN; k0 += 4) {
        const int ka = k0 + half * 2;
        v2f a, bm;
        // A 16x4 layout: lanes 0-15 -> K={k0,k0+1}, lanes 16-31 -> K={k0+2,k0+3}
        a.x = inB[(size_t)(ka    ) * FEAT + i0 + l15];
        a.y = inB[(size_t)(ka + 1) * FEAT + i0 + l15];
        // B 4x16 layout (mirrored): row = K, col = lane&15
        bm.x = lin_w[(size_t)(ka     + row_off) * DEMB + dc];
        bm.y = lin_w[(size_t)(ka + 1 + row_off) * DEMB + dc];
        c = __builtin_amdgcn_wmma_f32_16x16x4_f32(false, a, false, bm,
                                                  (short)0, c, false, false);
    }

    const float bias = (d < DEMB) ? lin_b[d] : 0.0f;
    float* dst = LR + ((size_t)b * FEAT + i0) * D2 + d;
#pragma unroll
    for (int r = 0; r < 8; ++r) {
        const int m = r + half * 8;    // C/D layout: VGPR r, lane half selects +8
        dst[(size_t)m * D2] = c[r] + bias;
    }
}

// ---------------------------------------------------------------------------
// Kernel 2: e[b,i,j] = sum_d leakyrelu(Lb[i,d] + R[j,d]) * a[d]; softmax over j.
// One 128-thread block per (b, i); thread t owns j = t.
// ---------------------------------------------------------------------------
__global__ __launch_bounds__(128) void gat_k2_scores(
    const float* __restrict__ LR,      // [B, FEAT, D2]
    const float* __restrict__ attn_w,  // [DEMB]
    float* __restrict__ attn)          // [B, FEAT, FEAT]
{
    __shared__ float sL[DEMB];
    __shared__ float sA[DEMB];
    __shared__ float sred[128];

    const int j = threadIdx.x;         // 0..127
    const int i = blockIdx.x;          // 0..127
    const int b = blockIdx.y;

    const float* Lrow = LR + ((size_t)b * FEAT + i) * D2;        // left+bias
    for (int dd = j; dd < DEMB; dd += 128) {
        sL[dd] = Lrow[dd];
        sA[dd] = attn_w[dd];
    }
    __syncthreads();

    const float* Rrow = LR + ((size_t)b * FEAT + j) * D2 + DEMB; // right half
    float acc = 0.0f;
    for (int dd = 0; dd < DEMB; dd += 4) {
        const float4 r4 = *(const float4*)(Rrow + dd);
        float p0 = sL[dd + 0] + r4.x;
        float p1 = sL[dd + 1] + r4.y;
        float p2 = sL[dd + 2] + r4.z;
        float p3 = sL[dd + 3] + r4.w;
        p0 = (p0 >= 0.0f) ? p0 : ALPHA * p0;
        p1 = (p1 >= 0.0f) ? p1 : ALPHA * p1;
        p2 = (p2 >= 0.0f) ? p2 : ALPHA * p2;
        p3 = (p3 >= 0.0f) ? p3 : ALPHA * p3;
        acc += p0 * sA[dd + 0] + p1 * sA[dd + 1] +
               p2 * sA[dd + 2] + p3 * sA[dd + 3];
    }

    // softmax over the 128 j's in this block
    sred[j] = acc;
    __syncthreads();
    for (int s = 64; s > 0; s >>= 1) {
        if (j < s) sred[j] = fmaxf(sred[j], sred[j + s]);
        __syncthreads();
    }
    const float mx = sred[0];
    __syncthreads();

    const float ex = __expf(acc - mx);
    sred[j] = ex;
    __syncthreads();
    for (int s = 64; s > 0; s >>= 1) {
        if (j < s) sred[j] += sred[j + s];
        __syncthreads();
    }
    const float inv = 1.0f / sred[0];

    attn[((size_t)b * FEAT + i) * FEAT + j] = ex * inv;
}

// ---------------------------------------------------------------------------
// Kernel 3: out[b, w, i] = sigmoid( sum_j attn[b,i,j] * input[b,w,j] )
// One wave per 16x16 tile, V_WMMA_F32_16X16X4_F32, K = 128 (j).
// ---------------------------------------------------------------------------
__global__ __launch_bounds__(32) void gat_k3_out(
    const float* __restrict__ attn,    // [B, FEAT, FEAT]
    const float* __restrict__ inp,     // [B, WN, FEAT]
    float* __restrict__ out)           // [B, WN, FEAT]
{
    const int lane = threadIdx.x;
    const int half = lane >> 4;
    const int l15  = lane & 15;
    const int n0   = blockIdx.x * 16;  // w tile in [0, 112)
    const int i0   = blockIdx.y * 16;  // node tile in [0, 128)
    const int b    = blockIdx.z;

    const int w = n0 + l15;
    const bool wok = (w < WN);

    const float* Ab  = attn + (size_t)b * FEAT * FEAT;
    const float* inB = inp  + (size_t)b * WN * FEAT;

    v8f c = {};
    for (int k0 = 0; k0 < FEAT; k0 += 4) {
        const int ka = k0 + half * 2;
        v2f a, bm;
        // A[m, k] = attn[b, i0+m, k]
        a.x = Ab[(size_t)(i0 + l15) * FEAT + ka];
        a.y = Ab[(size_t)(i0 + l15) * FEAT + ka + 1];
        // B[k, n] = x[b, j=k, w=n] = input[b, w, k]
        bm.x = wok ? inB[(size_t)w * FEAT + ka]     : 0.0f;
        bm.y = wok ? inB[(size_t)w * FEAT + ka + 1] : 0.0f;
        c = __builtin_amdgcn_wmma_f32_16x16x4_f32(false, a, false, bm,
                                                  (short)0, c, false, false);
    }

    if (wok) {
#pragma unroll
        for (int r = 0; r < 8; ++r) {
            const int m = r + half * 8;
            const float v = c[r];
            out[((size_t)b * WN + w) * FEAT + i0 + m] =
                1.0f / (1.0f + __expf(-v));
        }
    }
}

// ---------------------------------------------------------------------------
extern "C" void kernel_launch(void* const* d_in, const int* in_sizes, int n_in,
                              void* d_out, int out_size, void* d_ws, size_t ws_size,
                              hipStream_t stream) {
    const float* inp    = (const float*)d_in[0];  // [16,100,128]
    const float* lin_w  = (const float*)d_in[1];  // [200,200]
    const float* lin_b  = (const float*)d_in[2];  // [200]
    const float* attn_w = (const float*)d_in[3];  // [200]
    float* out = (float*)d_out;                   // [16,100,128]

    float* LR   = (float*)d_ws;                          // 16*128*400 floats
    float* attn = LR + (size_t)BATCH * FEAT * D2;        // 16*128*128 floats

    dim3 g1(D2 / 16, FEAT / 16, BATCH);       // (25, 8, 16)
    gat_k1_linear<<<g1, 32, 0, stream>>>(inp, lin_w, lin_b, LR);

    dim3 g2(FEAT, BATCH);                     // (128, 16)
    gat_k2_scores<<<g2, 128, 0, stream>>>(LR, attn_w, attn);

    dim3 g3((WN + 15) / 16, FEAT / 16, BATCH); // (7, 8, 16)
    gat_k3_out<<<g3, 32, 0, stream>>>(attn, inp, out);
}